// RoIReEmbed_49108656062898
// MI455X (gfx1250) — compile-verified
//
#include <hip/hip_runtime.h>
#include <hip/hip_bf16.h>
#include <math.h>

// ---------------------------------------------------------------------------
// RoIReEmbed pipeline for MI455X (gfx1250, wave32, WMMA).
// All 3x3 convs implemented as implicit GEMM on V_WMMA_F32_16X16X4_F32:
//   M = 16 out-channels (2 tiles for C=32), N = 16 output pixels,
//   K = Cin*3*3 = 288 in 72 steps of 4.
// Fragment layouts per CDNA5 ISA 7.12.2:
//   A 16x4 f32 : lane L -> m = L%16, VGPR v -> k = v + 2*(L/16)
//   B 4x16 f32 : lane L -> n = L%16, VGPR v -> k = v + 2*(L/16)
//   C/D 16x16  : lane L -> n = L%16, VGPR v -> m = v + 8*(L/16)
// ---------------------------------------------------------------------------

typedef __attribute__((ext_vector_type(2))) float v2f;
typedef __attribute__((ext_vector_type(8))) float v8f;

#define BN_SCALE 0.99999500003749981f   /* 1/sqrt(1+1e-5) */
#define LN_EPS   1e-5f

static __device__ inline v8f wmma_f32_k4(v2f a, v2f b, v8f c) {
  // (neg_a, A, neg_b, B, c_mod, C, reuse_a, reuse_b)
  return __builtin_amdgcn_wmma_f32_16x16x4_f32(false, a, false, b,
                                               (short)0, c, false, false);
}

// ---------------------------------------------------------------------------
// K1: gf = conv1x1(bn(gf)) + bias.  One thread per (b, pixel).
// ---------------------------------------------------------------------------
__global__ void k_gf_pre(const float* __restrict__ gf,
                         const float* __restrict__ bn_g,
                         const float* __restrict__ bn_b,
                         const float* __restrict__ w1,   // (32,32)
                         const float* __restrict__ b1,
                         float* __restrict__ y) {
  int idx = blockIdx.x * blockDim.x + threadIdx.x;      // 131072
  int b = idx >> 10, p = idx & 1023;
  const float* xb = gf + b * 32768 + p;
  float xin[32];
#pragma unroll
  for (int ci = 0; ci < 32; ++ci)
    xin[ci] = xb[ci * 1024] * (bn_g[ci] * BN_SCALE) + bn_b[ci];
  float* yb = y + b * 32768 + p;
  for (int co = 0; co < 32; ++co) {
    float acc = b1[co];
#pragma unroll
    for (int ci = 0; ci < 32; ++ci) acc += w1[co * 32 + ci] * xin[ci];
    yb[co * 1024] = acc;
  }
}

// ---------------------------------------------------------------------------
// K2: lf' = bn(lf) @ W^T + b; scatter-add into gf at (bidx,:,y,x).
// One thread per agent; duplicates handled by global_atomic_add_f32.
// ---------------------------------------------------------------------------
__global__ void k_lf_scatter(const float* __restrict__ lf,
                             const int* __restrict__ curr,
                             const float* __restrict__ bn_g,
                             const float* __restrict__ bn_b,
                             const float* __restrict__ w,   // (32,32)
                             const float* __restrict__ bias,
                             float* __restrict__ gfbuf) {
  int a = blockIdx.x * blockDim.x + threadIdx.x;        // 32768
  int b = a >> 8;                                       // BA/B = 256
  int yy = curr[2 * a], xx = curr[2 * a + 1];
  float xin[32];
#pragma unroll
  for (int ci = 0; ci < 32; ++ci)
    xin[ci] = lf[a * 32 + ci] * (bn_g[ci] * BN_SCALE) + bn_b[ci];
  for (int co = 0; co < 32; ++co) {
    float acc = bias[co];
#pragma unroll
    for (int ci = 0; ci < 32; ++ci) acc += w[co * 32 + ci] * xin[ci];
    atomicAdd(&gfbuf[((b * 32 + co) * 32 + yy) * 32 + xx], acc);
  }
}

// ---------------------------------------------------------------------------
// K3: per-batch mean / rstd over C*H*W = 32768 elements (population var).
// ---------------------------------------------------------------------------
__global__ void k_batch_stats(const float* __restrict__ x,
                              float* __restrict__ mean,
                              float* __restrict__ rstd) {
  __shared__ float s1[256], s2[256];
  int b = blockIdx.x, t = threadIdx.x;
  float a1 = 0.f, a2 = 0.f;
  for (int i = t; i < 32768; i += 256) {
    float v = x[b * 32768 + i];
    a1 += v; a2 += v * v;
  }
  s1[t] = a1; s2[t] = a2;
  __syncthreads();
  for (int s = 128; s > 0; s >>= 1) {
    if (t < s) { s1[t] += s1[t + s]; s2[t] += s2[t + s]; }
    __syncthreads();
  }
  if (t == 0) {
    float m = s1[0] * (1.f / 32768.f);
    float v = s2[0] * (1.f / 32768.f) - m * m;
    mean[b] = m;
    rstd[b] = rsqrtf(v + LN_EPS);
  }
}

// ---------------------------------------------------------------------------
// K4: elementwise LN apply with (C,H,W) weight/bias.
// ---------------------------------------------------------------------------
__global__ void k_ln_apply(const float* __restrict__ x,
                           const float* __restrict__ mean,
                           const float* __restrict__ rstd,
                           const float* __restrict__ lw,
                           const float* __restrict__ lb,
                           float* __restrict__ y) {
  int idx = blockIdx.x * blockDim.x + threadIdx.x;      // 4194304
  int b = idx >> 15, e = idx & 32767;
  y[idx] = (x[idx] - mean[b]) * rstd[b] * lw[e] + lb[e];
}

// ---------------------------------------------------------------------------
// K5: dilated 3x3 conv (pad=d) + bias + ReLU, via WMMA f32 16x16x4.
// One wave per (batch, image row): 2 N-tiles x 2 M-tiles = 4 WMMAs per
// k-step; A (weight) fragments are loaded once and reused for both N-tiles.
// ---------------------------------------------------------------------------
__global__ void k_gb_conv_wmma(const float* __restrict__ xin,
                               const float* __restrict__ w,   // (32,32,3,3)
                               const float* __restrict__ bias,
                               float* __restrict__ yout,
                               int dil) {
  const int lane  = threadIdx.x;        // 0..31
  const int b     = blockIdx.x >> 5;    // 128 batches
  const int row   = blockIdx.x & 31;    // 32 rows of 32 pixels
  const int nl    = lane & 15;
  const int khalf = lane >> 4;
  const int oy  = row;
  const int ox0 = nl, ox1 = nl + 16;
  const float* xb = xin + b * 32768;

  // weights are reused by all 4096 waves: hint them into cache (global_prefetch_b8)
  __builtin_prefetch(w + lane * 72, 0, 0);

  v8f acc00, acc01, acc10, acc11;       // [mtile][ntile]
#pragma unroll
  for (int v = 0; v < 8; ++v) {
    float b0 = bias[v + 8 * khalf];
    float b1 = bias[16 + v + 8 * khalf];
    acc00[v] = b0; acc01[v] = b0;
    acc10[v] = b1; acc11[v] = b1;
  }
  for (int ks = 0; ks < 72; ++ks) {
    v2f a0, a1, bf0, bf1;
#pragma unroll
    for (int v = 0; v < 2; ++v) {
      int k  = ks * 4 + v + 2 * khalf;          // k = ci*9 + ky*3 + kx
      int ci = k / 9, r = k - ci * 9;
      int ky = r / 3, kx = r - ky * 3;
      int iy  = oy + (ky - 1) * dil;
      int ix0 = ox0 + (kx - 1) * dil;
      int ix1 = ix0 + 16;
      bool rok = (unsigned)iy < 32u;
      const float* xrow = xb + ci * 1024 + iy * 32;
      bf0[v] = (rok && (unsigned)ix0 < 32u) ? xrow[ix0] : 0.f;
      bf1[v] = (rok && (unsigned)ix1 < 32u) ? xrow[ix1] : 0.f;
      a0[v] = w[nl * 288 + k];
      a1[v] = w[(nl + 16) * 288 + k];
    }
    acc00 = wmma_f32_k4(a0, bf0, acc00);
    acc10 = wmma_f32_k4(a1, bf0, acc10);
    acc01 = wmma_f32_k4(a0, bf1, acc01);
    acc11 = wmma_f32_k4(a1, bf1, acc11);
  }
  float* yb = yout + b * 32768 + oy * 32;
#pragma unroll
  for (int v = 0; v < 8; ++v) {
    int co = v + 8 * khalf;
    yb[co * 1024 + ox0]        = fmaxf(acc00[v], 0.f);
    yb[co * 1024 + ox1]        = fmaxf(acc01[v], 0.f);
    yb[(co + 16) * 1024 + ox0] = fmaxf(acc10[v], 0.f);
    yb[(co + 16) * 1024 + ox1] = fmaxf(acc11[v], 0.f);
  }
}

// ---------------------------------------------------------------------------
// K6: LN over the W axis only; weight/bias shape (32,) broadcast over W.
// One thread per (b,c,h) row.
// ---------------------------------------------------------------------------
__global__ void k_row_ln(const float* __restrict__ x,
                         const float* __restrict__ w,
                         const float* __restrict__ bb,
                         float* __restrict__ y) {
  int row = blockIdx.x * blockDim.x + threadIdx.x;      // 131072
  const float* xr = x + row * 32;
  float m = 0.f;
#pragma unroll
  for (int j = 0; j < 32; ++j) m += xr[j];
  m *= (1.f / 32.f);
  float v = 0.f;
#pragma unroll
  for (int j = 0; j < 32; ++j) { float d = xr[j] - m; v += d * d; }
  v *= (1.f / 32.f);
  float rs = rsqrtf(v + LN_EPS);
#pragma unroll
  for (int j = 0; j < 32; ++j)
    y[row * 32 + j] = (xr[j] - m) * rs * w[j] + bb[j];
}

// ---------------------------------------------------------------------------
// K7: patch head. One wave per agent: gather 32x11x11 patch (-1 outside),
// then 4 WMMA conv layers (bn folded into B-load) ping-ponged in LDS,
// final 3x3 -> 1x1 layer on VALU (no ReLU).
// ---------------------------------------------------------------------------
struct HeadParams {
  const float* bng[5];
  const float* bnb[5];
  const float* cw[5];
  const float* cb[5];
};

static __device__ void conv_layer_wmma(const float* __restrict__ wgt,
                                       const float* __restrict__ cb,
                                       const float* __restrict__ bng,
                                       const float* __restrict__ bnb,
                                       const float* in, float* out,
                                       int Sin, int lane) {
  const int Sout = Sin - 2;
  const int npix = Sout * Sout;
  const int ntiles = (npix + 15) >> 4;
  const int nl = lane & 15, khalf = lane >> 4;
  for (int nt = 0; nt < ntiles; ++nt) {
    int n = nt * 16 + nl;
    bool nvalid = n < npix;
    int oy = 0, ox = 0;
    if (nvalid) { oy = n / Sout; ox = n - oy * Sout; }
    v8f acc0, acc1;
#pragma unroll
    for (int v = 0; v < 8; ++v) {
      acc0[v] = cb[v + 8 * khalf];
      acc1[v] = cb[16 + v + 8 * khalf];
    }
    for (int ks = 0; ks < 72; ++ks) {
      v2f bf, a0, a1;
#pragma unroll
      for (int v = 0; v < 2; ++v) {
        int k  = ks * 4 + v + 2 * khalf;
        int ci = k / 9, r = k - ci * 9;
        int ky = r / 3, kx = r - ky * 3;
        float val = 0.f;
        if (nvalid) {
          float xv = in[ci * Sin * Sin + (oy + ky) * Sin + (ox + kx)];
          val = xv * (bng[ci] * BN_SCALE) + bnb[ci];   // per-channel BN
        }
        bf[v] = val;
        a0[v] = wgt[nl * 288 + k];
        a1[v] = wgt[(nl + 16) * 288 + k];
      }
      acc0 = wmma_f32_k4(a0, bf, acc0);
      acc1 = wmma_f32_k4(a1, bf, acc1);
    }
    if (nvalid) {
#pragma unroll
      for (int v = 0; v < 8; ++v) {
        int co = v + 8 * khalf;
        out[co * npix + n]        = fmaxf(acc0[v], 0.f);  // ReLU (layers 0..3)
        out[(co + 16) * npix + n] = fmaxf(acc1[v], 0.f);
      }
    }
    __syncthreads();
  }
  __syncthreads();
}

__global__ void k_patch_head(const float* __restrict__ gfn,
                             const int* __restrict__ curr,
                             HeadParams hp,
                             float* __restrict__ out) {
  __shared__ float bufA[32 * 11 * 11];   // 3872 floats
  __shared__ float bufB[32 * 9 * 9];     // 2592 floats
  const int a = blockIdx.x;              // 32768 agents, one wave each
  const int lane = threadIdx.x;          // 0..31 (EXEC all ones for WMMA)
  const int b = a >> 8;
  const int py0 = curr[2 * a] - 5, px0 = curr[2 * a + 1] - 5;

  // head weights are shared by every agent: prefetch (global_prefetch_b8)
  __builtin_prefetch(hp.cw[0] + lane * 72, 0, 0);
  __builtin_prefetch(hp.cw[1] + lane * 72, 0, 0);

  // gather 11x11x32 patch from the row-LN output; -1 outside the image
  for (int idx = lane; idx < 3872; idx += 32) {
    int c = idx / 121, rem = idx - c * 121;
    int py = rem / 11, px = rem - py * 11;
    int iy = py0 + py, ix = px0 + px;
    float v = -1.0f;
    if ((unsigned)iy < 32u && (unsigned)ix < 32u)
      v = gfn[((b * 32 + c) * 32 + iy) * 32 + ix];
    bufA[idx] = v;
  }
  __syncthreads();

  conv_layer_wmma(hp.cw[0], hp.cb[0], hp.bng[0], hp.bnb[0], bufA, bufB, 11, lane);
  conv_layer_wmma(hp.cw[1], hp.cb[1], hp.bng[1], hp.bnb[1], bufB, bufA,  9, lane);
  conv_layer_wmma(hp.cw[2], hp.cb[2], hp.bng[2], hp.bnb[2], bufA, bufB,  7, lane);
  conv_layer_wmma(hp.cw[3], hp.cb[3], hp.bng[3], hp.bnb[3], bufB, bufA,  5, lane);

  // layer 4: 3x3 -> 1x1, no ReLU. bufA holds 32x3x3 = 288 floats.
  {
    const float* w4 = hp.cw[4];
    const float* g4 = hp.bng[4];
    const float* b4 = hp.bnb[4];
    int co = lane;
    float acc = hp.cb[4][co];
    for (int ci = 0; ci < 32; ++ci) {
      float s = g4[ci] * BN_SCALE, bb = b4[ci];
#pragma unroll
      for (int t = 0; t < 9; ++t)
        acc += w4[co * 288 + ci * 9 + t] * (bufA[ci * 9 + t] * s + bb);
    }
    out[a * 32 + co] = acc;
  }
}

// ---------------------------------------------------------------------------
// Launch. Input order = setup_inputs() insertion order, params dict flattened
// in insertion order with lists expanded element-wise:
//  0 global_features 1 curr 2 target(unused) 3 local_features
//  4 gp_bn_g 5 gp_bn_b 6 gp_w 7 gp_b 8 lp_bn_g 9 lp_bn_b 10 lp_w 11 lp_b
//  12..17 gb_ln_w[6] 18..23 gb_ln_b[6] 24..29 gb_cw[6] 30..35 gb_cb[6]
//  36 gb_lnf_w 37 gb_lnf_b 38..42 bb_bn_g[5] 43..47 bb_bn_b[5]
//  48..52 bb_cw[5] 53..57 bb_cb[5]
// Workspace: 3 x 16 MiB gf buffers + stats (~50.4 MB).
// ---------------------------------------------------------------------------
extern "C" void kernel_launch(void* const* d_in, const int* in_sizes, int n_in,
                              void* d_out, int out_size, void* d_ws, size_t ws_size,
                              hipStream_t stream) {
  const float* gf   = (const float*)d_in[0];
  const int*   curr = (const int*)  d_in[1];
  const float* lf   = (const float*)d_in[3];
  const float* gp_bn_g = (const float*)d_in[4];
  const float* gp_bn_b = (const float*)d_in[5];
  const float* gp_w    = (const float*)d_in[6];
  const float* gp_b    = (const float*)d_in[7];
  const float* lp_bn_g = (const float*)d_in[8];
  const float* lp_bn_b = (const float*)d_in[9];
  const float* lp_w    = (const float*)d_in[10];
  const float* lp_b    = (const float*)d_in[11];
  const float* gb_ln_w[6]; const float* gb_ln_b[6];
  const float* gb_cw[6];   const float* gb_cb[6];
  for (int k = 0; k < 6; ++k) {
    gb_ln_w[k] = (const float*)d_in[12 + k];
    gb_ln_b[k] = (const float*)d_in[18 + k];
    gb_cw[k]   = (const float*)d_in[24 + k];
    gb_cb[k]   = (const float*)d_in[30 + k];
  }
  const float* gb_lnf_w = (const float*)d_in[36];
  const float* gb_lnf_b = (const float*)d_in[37];
  HeadParams hp;
  for (int k = 0; k < 5; ++k) {
    hp.bng[k] = (const float*)d_in[38 + k];
    hp.bnb[k] = (const float*)d_in[43 + k];
    hp.cw[k]  = (const float*)d_in[48 + k];
    hp.cb[k]  = (const float*)d_in[53 + k];
  }

  float* bufA = (float*)d_ws;
  float* bufB = bufA + 4194304;
  float* bufC = bufB + 4194304;
  float* meanp = bufC + 4194304;
  float* rstdp = meanp + 128;

  k_gf_pre<<<512, 256, 0, stream>>>(gf, gp_bn_g, gp_bn_b, gp_w, gp_b, bufA);
  k_lf_scatter<<<128, 256, 0, stream>>>(lf, curr, lp_bn_g, lp_bn_b, lp_w, lp_b, bufA);

  const int dils[6] = {1, 1, 1, 2, 4, 8};
  float* X = bufA; float* Y = bufB;
  for (int k = 0; k < 6; ++k) {
    k_batch_stats<<<128, 256, 0, stream>>>(X, meanp, rstdp);
    k_ln_apply<<<16384, 256, 0, stream>>>(X, meanp, rstdp, gb_ln_w[k], gb_ln_b[k], bufC);
    k_gb_conv_wmma<<<4096, 32, 0, stream>>>(bufC, gb_cw[k], gb_cb[k], Y, dils[k]);
    float* t = X; X = Y; Y = t;
  }

  k_row_ln<<<512, 256, 0, stream>>>(X, gb_lnf_w, gb_lnf_b, bufC);
  k_patch_head<<<32768, 32, 0, stream>>>(bufC, curr, hp, (float*)d_out);
}